// SelfAttention_70136815943710
// MI455X (gfx1250) — compile-verified
//
#include <hip/hip_runtime.h>

// ---------------------------------------------------------------------------
// Problem constants (from reference): B=2, S=2048, D=1024, H=16, hd=64
// ---------------------------------------------------------------------------
#define BATCH   2
#define SEQ     2048
#define DMODEL  1024
#define NHEADS  16
#define HEADD   64
#define BSROWS  (BATCH * SEQ)          // 4096

typedef float v2f __attribute__((ext_vector_type(2)));
typedef float v8f __attribute__((ext_vector_type(8)));

// fp32 tensor op: D(16x16) = A(16x4) * B(4x16) + C, full fp32 precision.
// 8-arg form: (neg_a, A, neg_b, B, c_mod, C, reuse_a, reuse_b)
static __device__ __forceinline__ v8f wmma4(v2f a, v2f b, v8f c) {
  return __builtin_amdgcn_wmma_f32_16x16x4_f32(false, a, false, b, (short)0, c,
                                               false, false);
}

static __device__ __forceinline__ v2f ld2(const float* p) {
  return *reinterpret_cast<const v2f*>(p);
}

// wave32 reductions across each 16-lane half (rows of the 16x16 C tile live
// in lanes 0-15 / 16-31 independently; xor masks < 16 stay inside a half).
static __device__ __forceinline__ float rowmax16(float v) {
#pragma unroll
  for (int m = 1; m < 16; m <<= 1) v = fmaxf(v, __shfl_xor(v, m, 32));
  return v;
}
static __device__ __forceinline__ float rowsum16(float v) {
#pragma unroll
  for (int m = 1; m < 16; m <<= 1) v += __shfl_xor(v, m, 32);
  return v;
}

// ---------------------------------------------------------------------------
// Generic TN GEMM: C[M,N] = A[M,K] * W[N,K]^T + bias[N]
//   A row-major with row stride lda; W row-major (N,K) stride K.
//   mode 0: C row-major, stride ldc.
//   mode 1: scatter into (B,H,S,hd) layout: n = h*64+d, m = b*S + s.
// WG = 128 threads (4 waves). Wave computes 32 rows x 64 cols of C.
// Software-pipelined K loop: fragments for k+4 are prefetched before the
// 8 WMMAs consuming the k fragments, hiding L0/L2 load latency.
// grid = (M/128, N/64)
// ---------------------------------------------------------------------------
__global__ void __launch_bounds__(128)
wmma_gemm_tn(const float* __restrict__ A, int lda,
             const float* __restrict__ W, int K,
             const float* __restrict__ bias,
             float* __restrict__ C, int ldc, int mode, int s_len) {
  const int wave = threadIdx.x >> 5;
  const int lane = threadIdx.x & 31;
  const int lrow = lane & 15;          // row/col within 16-tile
  const int lhi  = lane >> 4;          // 0 or 1 (selects K sub-pair)

  const int m0 = blockIdx.x * 128 + wave * 32;
  const int n0 = blockIdx.y * 64;

  v8f acc[2][4];
#pragma unroll
  for (int h = 0; h < 2; ++h)
#pragma unroll
    for (int t = 0; t < 4; ++t) acc[h][t] = (v8f)(0.0f);

  const float* a0p = A + (size_t)(m0 + lrow) * lda + lhi * 2;
  const float* a1p = A + (size_t)(m0 + 16 + lrow) * lda + lhi * 2;
  const float* w0p = W + (size_t)(n0 + 0  + lrow) * K + lhi * 2;
  const float* w1p = W + (size_t)(n0 + 16 + lrow) * K + lhi * 2;
  const float* w2p = W + (size_t)(n0 + 32 + lrow) * K + lhi * 2;
  const float* w3p = W + (size_t)(n0 + 48 + lrow) * K + lhi * 2;

  // software pipeline: preload k=0 fragments
  v2f a0 = ld2(a0p), a1 = ld2(a1p);
  v2f b0 = ld2(w0p), b1 = ld2(w1p), b2 = ld2(w2p), b3 = ld2(w3p);

  auto fma8 = [&](v2f A0, v2f A1, v2f B0, v2f B1, v2f B2, v2f B3) {
    acc[0][0] = wmma4(A0, B0, acc[0][0]);
    acc[1][0] = wmma4(A1, B0, acc[1][0]);
    acc[0][1] = wmma4(A0, B1, acc[0][1]);
    acc[1][1] = wmma4(A1, B1, acc[1][1]);
    acc[0][2] = wmma4(A0, B2, acc[0][2]);
    acc[1][2] = wmma4(A1, B2, acc[1][2]);
    acc[0][3] = wmma4(A0, B3, acc[0][3]);
    acc[1][3] = wmma4(A1, B3, acc[1][3]);
  };

#pragma unroll 8
  for (int k = 0; k < K - 4; k += 4) {
    // prefetch fragments for k+4 (independent of the WMMAs below)
    v2f na0 = ld2(a0p + k + 4);
    v2f na1 = ld2(a1p + k + 4);
    v2f nb0 = ld2(w0p + k + 4);
    v2f nb1 = ld2(w1p + k + 4);
    v2f nb2 = ld2(w2p + k + 4);
    v2f nb3 = ld2(w3p + k + 4);
    fma8(a0, a1, b0, b1, b2, b3);
    a0 = na0; a1 = na1;
    b0 = nb0; b1 = nb1; b2 = nb2; b3 = nb3;
  }
  fma8(a0, a1, b0, b1, b2, b3);   // last K-step

  // epilogue: bias + store.  C tile layout: VGPR i -> row (i + lhi*8),
  // col = lrow within each 16x16 tile.
#pragma unroll
  for (int t = 0; t < 4; ++t) {
    const int n = n0 + t * 16 + lrow;
    const float bv = bias[n];
#pragma unroll
    for (int h = 0; h < 2; ++h) {
#pragma unroll
      for (int i = 0; i < 8; ++i) {
        const int m = m0 + h * 16 + lhi * 8 + i;
        const float val = acc[h][t][i] + bv;
        if (mode == 0) {
          C[(size_t)m * ldc + n] = val;
        } else {
          const int b = m / s_len, s = m % s_len;
          const int hh = n >> 6, d = n & 63;
          C[(((size_t)b * NHEADS + hh) * s_len + s) * HEADD + d] = val;
        }
      }
    }
  }
}

// ---------------------------------------------------------------------------
// RoPE, in place on the q and k thirds of the fused qkv buffer (4096 x 3072).
// One thread handles one (val, partner) rotation pair.
// ---------------------------------------------------------------------------
__global__ void __launch_bounds__(256)
rope_kernel(float* __restrict__ qkv) {
  const int idx = blockIdx.x * blockDim.x + threadIdx.x;
  // total = BSROWS * 2(qk) * NHEADS * 32(pairs) = 4096*1024
  if (idx >= BSROWS * 1024) return;
  const int j   = idx & 31;
  const int h   = (idx >> 5) & 15;
  const int qk  = (idx >> 9) & 1;
  const int row = idx >> 10;                // 0..4095
  const int pos = row & (SEQ - 1);          // s position

  const float inv = powf(10000.0f, -(float)(2 * j) / (float)HEADD);
  const float ang = (float)pos * inv;
  const float cs = cosf(ang);
  const float sn = sinf(ang);

  const size_t base = (size_t)row * (3 * DMODEL) + qk * DMODEL + h * HEADD + j;
  const float t1 = qkv[base];
  const float t2 = qkv[base + 32];
  qkv[base]      = t1 * cs - t2 * sn;   // d < 32: t*cos - t2*sin
  qkv[base + 32] = t2 * cs + t1 * sn;   // d >=32: t*cos + t1*sin
}

// ---------------------------------------------------------------------------
// Flash-style attention.  Q/K/V in (B*H, S, 64) layout, fp32.
// WG = 128 threads (4 waves); wave owns 16 query rows; WG covers 64 rows.
// Streams over keys in tiles of 16 with online softmax.
// grid = (S/64, B*H)
// ---------------------------------------------------------------------------
#define KPAD 68   // LDS row pitch (floats): conflict-free + 16B aligned

__global__ void __launch_bounds__(128)
attn_kernel(const float* __restrict__ Q, const float* __restrict__ K,
            const float* __restrict__ V, float* __restrict__ ctx) {
  __shared__ float Klds[16 * KPAD];
  __shared__ float Vlds[16 * KPAD];
  __shared__ float pbuf[4][16 * 16];

  const int tid  = threadIdx.x;
  const int wave = tid >> 5;
  const int lane = tid & 31;
  const int lrow = lane & 15;
  const int lhi  = lane >> 4;

  const int bh = blockIdx.y;
  const int b  = bh >> 4;
  const int h  = bh & 15;
  const int m0 = blockIdx.x * 64 + wave * 16;

  const float* Qb = Q + (size_t)bh * SEQ * HEADD;
  const float* Kb = K + (size_t)bh * SEQ * HEADD;
  const float* Vb = V + (size_t)bh * SEQ * HEADD;

  // Q fragment in WMMA A-layout, pre-scaled by 1/sqrt(hd) = 0.125
  v2f qa[16];
  {
    const float* qp = Qb + (size_t)(m0 + lrow) * HEADD + lhi * 2;
#pragma unroll
    for (int kk = 0; kk < 16; ++kk) {
      v2f t = ld2(qp + kk * 4);
      qa[kk] = t * 0.125f;
    }
  }

  v8f acc[4];
#pragma unroll
  for (int t = 0; t < 4; ++t) acc[t] = (v8f)(0.0f);
  float mrow[8], lsum[8];
#pragma unroll
  for (int i = 0; i < 8; ++i) { mrow[i] = -3.0e38f; lsum[i] = 0.0f; }

  const int key0 = (tid >> 3);      // 0..15 key row for coop load
  const int seg  = (tid & 7) * 8;   // 8-float segment

  for (int kt = 0; kt < SEQ / 16; ++kt) {
    __syncthreads();  // protect previous tile reads
    {
      const float* kp = Kb + (size_t)(kt * 16 + key0) * HEADD + seg;
      const float* vp = Vb + (size_t)(kt * 16 + key0) * HEADD + seg;
      float4 k0 = *(const float4*)(kp);
      float4 k1 = *(const float4*)(kp + 4);
      float4 v0 = *(const float4*)(vp);
      float4 v1 = *(const float4*)(vp + 4);
      *(float4*)&Klds[key0 * KPAD + seg]     = k0;
      *(float4*)&Klds[key0 * KPAD + seg + 4] = k1;
      *(float4*)&Vlds[key0 * KPAD + seg]     = v0;
      *(float4*)&Vlds[key0 * KPAD + seg + 4] = v1;
    }
    __syncthreads();

    // ---- scores: S(16x16) = Qhat(16x64) * K(16keys x 64)^T ----
    v8f sc = (v8f)(0.0f);
#pragma unroll
    for (int kk = 0; kk < 16; ++kk) {
      v2f kb = ld2(&Klds[lrow * KPAD + kk * 4 + lhi * 2]);
      sc = wmma4(qa[kk], kb, sc);
    }

    // ---- online softmax update ----
    float corr[8];
#pragma unroll
    for (int i = 0; i < 8; ++i) {
      float v  = sc[i];
      float rm = rowmax16(v);
      float mn = fmaxf(mrow[i], rm);
      float p  = __expf(v - mn);
      float rs = rowsum16(p);
      corr[i]  = __expf(mrow[i] - mn);
      lsum[i]  = lsum[i] * corr[i] + rs;
      mrow[i]  = mn;
      sc[i]    = p;
    }
#pragma unroll
    for (int t = 0; t < 4; ++t)
#pragma unroll
      for (int i = 0; i < 8; ++i) acc[t][i] *= corr[i];

    // ---- transpose P (D-layout -> A-layout) through per-wave LDS ----
    float* pb = pbuf[wave];
#pragma unroll
    for (int i = 0; i < 8; ++i) pb[(i + lhi * 8) * 16 + lrow] = sc[i];
    asm volatile("s_wait_dscnt 0" ::: "memory");

    v2f pa[4];
#pragma unroll
    for (int kk = 0; kk < 4; ++kk)
      pa[kk] = ld2(&pb[lrow * 16 + kk * 4 + lhi * 2]);

    // ---- O(16x64) += P(16x16) * V(16keys x 64) ----
#pragma unroll
    for (int ng = 0; ng < 4; ++ng) {
#pragma unroll
      for (int kk = 0; kk < 4; ++kk) {
        const int kr = kk * 4 + lhi * 2;
        v2f vb;
        vb.x = Vlds[(kr + 0) * KPAD + ng * 16 + lrow];
        vb.y = Vlds[(kr + 1) * KPAD + ng * 16 + lrow];
        acc[ng] = wmma4(pa[kk], vb, acc[ng]);
      }
    }
  }

  // ---- epilogue: normalize and scatter to ctx (B, S, H*hd) row-major ----
#pragma unroll
  for (int ng = 0; ng < 4; ++ng) {
#pragma unroll
    for (int i = 0; i < 8; ++i) {
      const int s = m0 + lhi * 8 + i;
      const int d = h * HEADD + ng * 16 + lrow;
      ctx[((size_t)b * SEQ + s) * DMODEL + d] = acc[ng][i] / lsum[i];
    }
  }
}

// ---------------------------------------------------------------------------
// Host-side launcher
// ---------------------------------------------------------------------------
extern "C" void kernel_launch(void* const* d_in, const int* in_sizes, int n_in,
                              void* d_out, int out_size, void* d_ws,
                              size_t ws_size, hipStream_t stream) {
  (void)in_sizes; (void)n_in; (void)out_size; (void)ws_size;

  const float* x    = (const float*)d_in[0];   // (2, 2048, 1024)
  const float* wIn  = (const float*)d_in[1];   // (3072, 1024)
  const float* bIn  = (const float*)d_in[2];   // (3072,)
  const float* wOut = (const float*)d_in[3];   // (1024, 1024)
  const float* bOut = (const float*)d_in[4];   // (1024,)
  float* out = (float*)d_out;                  // (2, 2048, 1024)

  // workspace layout (floats)
  float* qkv = (float*)d_ws;                          // 4096 x 3072
  float* Qb  = qkv + (size_t)BSROWS * 3 * DMODEL;     // (BH, S, 64)
  float* Kb  = Qb + (size_t)BSROWS * DMODEL;
  float* Vb  = Kb + (size_t)BSROWS * DMODEL;
  float* ctx = Vb + (size_t)BSROWS * DMODEL;          // 4096 x 1024

  const dim3 blk(128);

  // 1) fused QKV in-projection: qkv = x * in_proj_w^T + in_proj_b
  wmma_gemm_tn<<<dim3(BSROWS / 128, (3 * DMODEL) / 64), blk, 0, stream>>>(
      x, DMODEL, wIn, DMODEL, bIn, qkv, 3 * DMODEL, /*mode=*/0, SEQ);

  // 2) RoPE in place on q and k thirds
  rope_kernel<<<(BSROWS * 1024 + 255) / 256, 256, 0, stream>>>(qkv);

  // 3) second projections -> (B,H,S,hd) layouts
  wmma_gemm_tn<<<dim3(BSROWS / 128, DMODEL / 64), blk, 0, stream>>>(
      qkv + 0 * DMODEL, 3 * DMODEL, wIn + 0 * (size_t)DMODEL * DMODEL, DMODEL,
      bIn + 0 * DMODEL, Qb, DMODEL, /*mode=*/1, SEQ);
  wmma_gemm_tn<<<dim3(BSROWS / 128, DMODEL / 64), blk, 0, stream>>>(
      qkv + 1 * DMODEL, 3 * DMODEL, wIn + 1 * (size_t)DMODEL * DMODEL, DMODEL,
      bIn + 1 * DMODEL, Kb, DMODEL, /*mode=*/1, SEQ);
  wmma_gemm_tn<<<dim3(BSROWS / 128, DMODEL / 64), blk, 0, stream>>>(
      qkv + 2 * DMODEL, 3 * DMODEL, wIn + 2 * (size_t)DMODEL * DMODEL, DMODEL,
      bIn + 2 * DMODEL, Vb, DMODEL, /*mode=*/1, SEQ);

  // 4) attention (flash-style, online softmax)
  attn_kernel<<<dim3(SEQ / 64, BATCH * NHEADS), blk, 0, stream>>>(Qb, Kb, Vb,
                                                                  ctx);

  // 5) output projection: out = ctx * out_proj_w^T + out_proj_b
  wmma_gemm_tn<<<dim3(BSROWS / 128, DMODEL / 64), blk, 0, stream>>>(
      ctx, DMODEL, wOut, DMODEL, bOut, out, DMODEL, /*mode=*/0, SEQ);
}